// MultiEncoderYawModel_8761733284272
// MI455X (gfx1250) — compile-verified
//
#include <hip/hip_runtime.h>
#include <cstdint>

typedef __attribute__((ext_vector_type(2))) float v2f;
typedef __attribute__((ext_vector_type(4))) float f4;
typedef __attribute__((ext_vector_type(8))) float v8f;

#define D_DIM 256   // feature dim
#define L_DIM 2     // latent dim
#define E_DIM 64    // experts

// One wave processes 16 consecutive rows (2 rows per iteration, 16 lanes each);
// block = 8 waves = 128 rows.
__global__ __launch_bounds__(256) void multi_encoder_yaw_kernel(
    const float* __restrict__ x,       // [N, D]
    const int*   __restrict__ eidx,    // [N]
    const float* __restrict__ encW,    // [E, L, D]
    const float* __restrict__ encB,    // [E, L]
    const float* __restrict__ decW,    // [1, L]
    const float* __restrict__ decB,    // [1]
    float* __restrict__ out,           // [N] y then [N,2] z
    int N)
{
    extern __shared__ float lds_w[];   // E*L*D floats = 128 KB

    const int tid  = threadIdx.x;
    const int lane = tid & 31;
    const int wave = tid >> 5;
    const int sl   = lane & 15;        // sub-lane within half-wave
    const int half = lane >> 4;        // 0: even row, 1: odd row

    // ---- Stage all expert weights into LDS with CDNA5 async-to-LDS DMA ----
    // 32768 floats = 8192 float4 chunks; 256 threads -> 32 chunks per thread.
    for (int c = 0; c < 32; ++c) {
        const int f4i = c * 256 + tid;
        const uint32_t ldsoff = (uint32_t)(uintptr_t)(&lds_w[f4i * 4]);
        const uint64_t gaddr  = (uint64_t)(uintptr_t)(encW + (size_t)f4i * 4);
        asm volatile("global_load_async_to_lds_b128 %0, %1, off"
                     :: "v"(ldsoff), "v"(gaddr)
                     : "memory");
    }
    asm volatile("s_wait_asynccnt 0" ::: "memory");
    __syncthreads();

    // Decoder params are uniform -> scalar loads.
    const float dw0 = decW[0];
    const float dw1 = decW[1];
    const float db  = decB[0];

    float* outY = out;
    float* outZ = out + N;

    const int base = blockIdx.x * 128 + wave * 16;

    // WMMA A-tile: lanes 0..15 row M=lane hold K0=z0, K1=z1; lanes 16..31 = 0.
    v2f a = {0.0f, 0.0f};

    #pragma unroll
    for (int i = 0; i < 8; ++i) {
        const int row   = base + 2 * i + half;   // half 0 -> even, half 1 -> odd
        const bool valid = row < N;
        const int rr    = valid ? row : 0;

        const int e = eidx[rr];                  // uniform per half-wave

        // x row: 16 lanes cover D=256; lane handles elements {j*64 + 4*sl .. +3}
        const f4* xr = (const f4*)(x + (size_t)rr * D_DIM);
        const f4 x0 = xr[sl];
        const f4 x1 = xr[sl + 16];
        const f4 x2 = xr[sl + 32];
        const f4 x3 = xr[sl + 48];

        // weights from LDS, layout [e][l][d] (f4 granularity: 64 per l)
        const f4* w = (const f4*)(lds_w + (size_t)e * (L_DIM * D_DIM));
        const f4 a0 = w[sl];      const f4 a1 = w[sl + 16];
        const f4 a2 = w[sl + 32]; const f4 a3 = w[sl + 48];
        const f4 b0 = w[sl + 64]; const f4 b1 = w[sl + 80];
        const f4 b2 = w[sl + 96]; const f4 b3 = w[sl + 112];

        float s0 = x0.x*a0.x + x0.y*a0.y + x0.z*a0.z + x0.w*a0.w
                 + x1.x*a1.x + x1.y*a1.y + x1.z*a1.z + x1.w*a1.w
                 + x2.x*a2.x + x2.y*a2.y + x2.z*a2.z + x2.w*a2.w
                 + x3.x*a3.x + x3.y*a3.y + x3.z*a3.z + x3.w*a3.w;
        float s1 = x0.x*b0.x + x0.y*b0.y + x0.z*b0.z + x0.w*b0.w
                 + x1.x*b1.x + x1.y*b1.y + x1.z*b1.z + x1.w*b1.w
                 + x2.x*b2.x + x2.y*b2.y + x2.z*b2.z + x2.w*b2.w
                 + x3.x*b3.x + x3.y*b3.y + x3.z*b3.z + x3.w*b3.w;

        // Butterfly within each 16-lane half -> all 16 lanes hold the row total.
        #pragma unroll
        for (int m = 8; m >= 1; m >>= 1) {
            s0 += __shfl_xor(s0, m, 32);
            s1 += __shfl_xor(s1, m, 32);
        }

        const float z0 = s0 + encB[e * 2 + 0];
        const float z1 = s1 + encB[e * 2 + 1];

        // One lane per half-wave writes this row's z.
        if (valid && sl == 0) {
            float2 zz; zz.x = z0; zz.y = z1;
            *(float2*)(outZ + (size_t)row * 2) = zz;
        }

        // Cross-half exchange: every lane gets the other row's z too.
        const float t0 = __shfl_xor(z0, 16, 32);
        const float t1 = __shfl_xor(z1, 16, 32);

        // Lane m (<16) needs row base+m. Even lane -> own (even row, half 0
        // source); odd lane -> partner (odd row). Lanes 16..31 never match.
        const float sel0 = (lane & 1) ? t0 : z0;
        const float sel1 = (lane & 1) ? t1 : z1;
        if ((lane >> 1) == i) { a.x = sel0; a.y = sel1; }
    }

    // ---- Decoder via V_WMMA_F32_16X16X4_F32 ----
    // B[k,n]: K0 row = dw0 (lanes 0-15), K1 row = dw1; K2,K3 (lanes 16-31) = 0.
    v2f b;
    b.x = (half == 0) ? dw0 : 0.0f;
    b.y = (half == 0) ? dw1 : 0.0f;
    v8f c = {db, db, db, db, db, db, db, db};

    // D[m,n] = z0[m]*dw0 + z1[m]*dw1 + db = y[m] (identical across n)
    v8f d = __builtin_amdgcn_wmma_f32_16x16x4_f32(
        /*neg_a=*/false, a, /*neg_b=*/false, b,
        /*c_mod=*/(short)0, c, /*reuse_a=*/false, /*reuse_b=*/false);

    // C/D layout: lane 0 (N=0) holds M=0..7 in VGPR0..7; lane 16 holds M=8..15.
    if (base + 16 <= N) {
        // Full tile: vector stores (outY+base is 64B aligned).
        if (lane == 0) {
            f4 lo; lo.x = d[0]; lo.y = d[1]; lo.z = d[2]; lo.w = d[3];
            f4 hi; hi.x = d[4]; hi.y = d[5]; hi.z = d[6]; hi.w = d[7];
            *(f4*)(outY + base)     = lo;
            *(f4*)(outY + base + 4) = hi;
        } else if (lane == 16) {
            f4 lo; lo.x = d[0]; lo.y = d[1]; lo.z = d[2]; lo.w = d[3];
            f4 hi; hi.x = d[4]; hi.y = d[5]; hi.z = d[6]; hi.w = d[7];
            *(f4*)(outY + base + 8)  = lo;
            *(f4*)(outY + base + 12) = hi;
        }
    } else {
        if (lane == 0) {
            for (int m = 0; m < 8; ++m) {
                const int rowm = base + m;
                if (rowm < N) outY[rowm] = d[m];
            }
        } else if (lane == 16) {
            for (int m = 0; m < 8; ++m) {
                const int rowm = base + 8 + m;
                if (rowm < N) outY[rowm] = d[m];
            }
        }
    }
}

extern "C" void kernel_launch(void* const* d_in, const int* in_sizes, int n_in,
                              void* d_out, int out_size, void* d_ws, size_t ws_size,
                              hipStream_t stream) {
    const float* x    = (const float*)d_in[0];
    const int*   idx  = (const int*)d_in[1];
    const float* encW = (const float*)d_in[2];
    const float* encB = (const float*)d_in[3];
    const float* decW = (const float*)d_in[4];
    const float* decB = (const float*)d_in[5];
    float* out = (float*)d_out;

    const int N = in_sizes[1];                 // individual_idx element count
    const int blocks = (N + 127) / 128;        // 128 rows per block
    const size_t shmem = (size_t)E_DIM * L_DIM * D_DIM * sizeof(float); // 128 KB

    // Allow >64KB dynamic LDS (WGP supports up to 320 KB). Deterministic,
    // not a stream op, safe under graph capture.
    (void)hipFuncSetAttribute((const void*)multi_encoder_yaw_kernel,
                              hipFuncAttributeMaxDynamicSharedMemorySize,
                              (int)shmem);

    multi_encoder_yaw_kernel<<<blocks, 256, shmem, stream>>>(
        x, idx, encW, encB, decW, decB, out, N);
}